// Balanced_sinkhorn_ce_75411035783872
// MI455X (gfx1250) — compile-verified
//
#include <hip/hip_runtime.h>
#include <math.h>

// ---------------------------------------------------------------------------
// MI455X (gfx1250) implementation of Balanced Sinkhorn CE.
//
// Roofline: dominant work is the 8192x2048x256 f32 GEMM (8.6 GFLOP, 66 MB
// traffic) -> matrix-core bound -> V_WMMA_F32_16X16X4_F32 (f32 needed because
// logits feed exp(x*20)). The GEMM register-blocks 2x2 WMMA tiles per wave
// (32x32 output) so each A/B fragment feeds two matrix ops: 1.5 loads/WMMA.
// Everything downstream (E, log_prob: 3 x 8 MB) is L2-resident (192 MB L2),
// done as matvec reductions with f64 accumulation. Sinkhorn is kept in
// factored form Q = diag(r) E diag(s); forward and the hand-derived
// reverse-mode (for grad wrt w) are each 2 matvecs / iteration.
// ---------------------------------------------------------------------------

#define BSZ 8192
#define DF  2048
#define CC  256
#define NOUT 10

typedef __attribute__((ext_vector_type(2))) float v2f;
typedef __attribute__((ext_vector_type(8))) float v8f;

// ---------------- reduction helpers (256-thread blocks) --------------------
__device__ __forceinline__ double blk_sum(double v, double* sm) {
  int c = threadIdx.x; sm[c] = v; __syncthreads();
  for (int s = 128; s; s >>= 1) { if (c < s) sm[c] += sm[c + s]; __syncthreads(); }
  double r = sm[0]; __syncthreads();
  return r;
}
__device__ __forceinline__ double blk_max(double v, double* sm) {
  int c = threadIdx.x; sm[c] = v; __syncthreads();
  for (int s = 128; s; s >>= 1) { if (c < s) sm[c] = fmax(sm[c], sm[c + s]); __syncthreads(); }
  double r = sm[0]; __syncthreads();
  return r;
}

// ---------------- stage 0: norms ------------------------------------------
__global__ void k_rownorm(const float* __restrict__ F, float* __restrict__ rn) {
  int b = blockIdx.x;
  float acc = 0.f;
  for (int d = threadIdx.x; d < DF; d += 256) { float x = F[(size_t)b * DF + d]; acc += x * x; }
  __shared__ float sm[256];
  sm[threadIdx.x] = acc; __syncthreads();
  for (int s = 128; s; s >>= 1) { if (threadIdx.x < s) sm[threadIdx.x] += sm[threadIdx.x + s]; __syncthreads(); }
  if (!threadIdx.x) rn[b] = sqrtf(sm[0]);
}

__global__ void k_colnorm(const float* __restrict__ H, float* __restrict__ cn) {
  int c = blockIdx.x;
  float acc = 0.f;
  for (int d = threadIdx.x; d < DF; d += 256) { float x = H[(size_t)d * CC + c]; acc += x * x; }
  __shared__ float sm[256];
  sm[threadIdx.x] = acc; __syncthreads();
  for (int s = 128; s; s >>= 1) { if (threadIdx.x < s) sm[threadIdx.x] += sm[threadIdx.x + s]; __syncthreads(); }
  if (!threadIdx.x) cn[c] = sqrtf(sm[0]);
}

// ---------------- stage 1: WMMA f32 GEMM, 2x2 register-blocked -------------
// Wave computes a 32x32 output patch as 2x2 WMMA 16x16 tiles; K-loop step 4
// via V_WMMA_F32_16X16X4_F32. Fragment layouts (wave32):
//   A 16x4: lanes 0-15 hold K=0,1 (VGPR0,1); lanes 16-31 hold K=2,3.
//   B 4x16: rows K=0,1 in lanes 0-15; K=2,3 in lanes 16-31; N = lane%16.
//   C/D 16x16: VGPR j holds M=j (lanes 0-15) / M=j+8 (lanes 16-31), N=lane%16.
// Each A frag feeds 2 WMMAs and each B frag feeds 2 WMMAs: 1.5 loads/WMMA.
__global__ __launch_bounds__(256) void k_gemm(const float* __restrict__ F,
                                              const float* __restrict__ H,
                                              const float* __restrict__ rn,
                                              const float* __restrict__ cn,
                                              float* __restrict__ logits) {
  const int lane = threadIdx.x & 31;
  const int wave = threadIdx.x >> 5;
  const int l = lane & 15;
  const int h = lane >> 4;
  const int m0 = blockIdx.x * 64 + (wave >> 2) * 32;   // 2 row-tiles of 16
  const int n0 = blockIdx.y * 128 + (wave & 3) * 32;   // 2 col-tiles of 16
  const float* arow0 = F + (size_t)(m0 + l) * DF;
  const float* arow1 = F + (size_t)(m0 + 16 + l) * DF;
  const int c0 = n0 + l;
  const int c1 = n0 + 16 + l;

  v8f acc00 = {}, acc01 = {}, acc10 = {}, acc11 = {};
#pragma unroll 4
  for (int k = 0; k < DF; k += 4) {
    if ((k & 255) == 0) {                      // global_prefetch_b8, WGP scope
      __builtin_prefetch(arow0 + k + 512, 0, 3);
      __builtin_prefetch(arow1 + k + 512, 0, 3);
    }
    v2f a0 = *(const v2f*)(arow0 + k + 2 * h);
    v2f a1 = *(const v2f*)(arow1 + k + 2 * h);
    v2f b0, b1;
    b0[0] = H[(size_t)(k + 2 * h) * CC + c0];
    b0[1] = H[(size_t)(k + 2 * h + 1) * CC + c0];
    b1[0] = H[(size_t)(k + 2 * h) * CC + c1];
    b1[1] = H[(size_t)(k + 2 * h + 1) * CC + c1];
    acc00 = __builtin_amdgcn_wmma_f32_16x16x4_f32(false, a0, false, b0, (short)0, acc00, false, false);
    acc01 = __builtin_amdgcn_wmma_f32_16x16x4_f32(false, a0, false, b1, (short)0, acc01, false, false);
    acc10 = __builtin_amdgcn_wmma_f32_16x16x4_f32(false, a1, false, b0, (short)0, acc10, false, false);
    acc11 = __builtin_amdgcn_wmma_f32_16x16x4_f32(false, a1, false, b1, (short)0, acc11, false, false);
  }
  float icn0 = 1.0f / cn[c0];
  float icn1 = 1.0f / cn[c1];
#pragma unroll
  for (int j = 0; j < 8; j++) {
    int r0 = m0 + j + 8 * h;
    int r1 = m0 + 16 + j + 8 * h;
    float irn0 = 1.0f / rn[r0];
    float irn1 = 1.0f / rn[r1];
    logits[(size_t)r0 * CC + c0] = acc00[j] * irn0 * icn0;
    logits[(size_t)r0 * CC + c1] = acc01[j] * irn0 * icn1;
    logits[(size_t)r1 * CC + c0] = acc10[j] * irn1 * icn0;
    logits[(size_t)r1 * CC + c1] = acc11[j] * irn1 * icn1;
  }
}

// ---------------- stage 2: log_softmax(logits/0.1) and E=exp(logits/0.05) --
__global__ void k_lse(const float* __restrict__ logits, float* __restrict__ E,
                      float* __restrict__ L) {
  int b = blockIdx.x, c = threadIdx.x;
  float x = logits[(size_t)b * CC + c];
  double z = (double)x * 10.0;
  __shared__ double sm[256];
  double M = blk_max(z, sm);
  double S = blk_sum(exp(z - M), sm);
  L[(size_t)b * CC + c] = (float)(z - M - log(S));
  E[(size_t)b * CC + c] = (float)exp((double)x * 20.0);  // global-sum normalization cancels
}

// ---------------- matvec reductions over E (optionally * L) ----------------
// out[c] = sum_b E[b,c] (*L[b,c]) * x[b]
__global__ void k_colmv(const float* __restrict__ E, const float* __restrict__ L,
                        int useL, const double* __restrict__ x, double* __restrict__ out) {
  int c = blockIdx.x;
  double acc = 0.0;
  for (int b = threadIdx.x; b < BSZ; b += 256) {
    double e = (double)E[(size_t)b * CC + c];
    if (useL) e *= (double)L[(size_t)b * CC + c];
    acc += e * x[b];
  }
  __shared__ double sm[256];
  double r = blk_sum(acc, sm);
  if (!threadIdx.x) out[c] = r;
}

// out[b] = f( sum_c E[b,c] (*L[b,c]) * y[c] );  mode 1: f(v)=1/(B*v)
__global__ void k_rowmv(const float* __restrict__ E, const float* __restrict__ L,
                        int useL, const double* __restrict__ y, double* __restrict__ out,
                        int mode) {
  int lane = threadIdx.x & 31;
  int b = blockIdx.x * 8 + (threadIdx.x >> 5);
  double acc = 0.0;
  for (int c = lane; c < CC; c += 32) {
    double e = (double)E[(size_t)b * CC + c];
    if (useL) e *= (double)L[(size_t)b * CC + c];
    acc += e * y[c];
  }
  for (int off = 16; off; off >>= 1) acc += __shfl_down(acc, off, 32);
  if (!lane) out[b] = (mode == 1) ? 1.0 / ((double)BSZ * acc) : acc;
}

// ---------------- small per-iteration kernels ------------------------------
__global__ void k_pre(const double* __restrict__ w, double* __restrict__ p,
                      double* __restrict__ pbar, double* __restrict__ reg) {
  int c = threadIdx.x;
  __shared__ double sm[256];
  double wc = w[c];
  double M = blk_max(wc, sm);
  double S = blk_sum(exp(wc - M), sm);
  double pc = exp(wc - M) / S;
  p[c] = pc;
  pbar[c] = 0.0;
  double logp = wc - M - log(S);
  double R = blk_sum(-log((double)CC) - logp, sm);
  if (!c) reg[0] = R / ((double)CC * (double)CC);
}

__global__ void k_r(const double* p, const double* u, double* r) {
  int c = threadIdx.x;
  r[c] = p[c] / u[c];
}

// vbar[b] = coef * s[b]^2 * sb[b]   (coef = +B for k=K with sb=q, -B otherwise)
__global__ void k_vbar(const double* s, const double* sb, double coef, double* vbar) {
  int i = blockIdx.x * 256 + threadIdx.x;
  vbar[i] = coef * s[i] * s[i] * sb[i];
}

__global__ void k_bkc(const double* cacc, const double* t, int isLast,
                      const double* u, const double* r, double* pbar, double* ubar) {
  int c = threadIdx.x;
  double rb = cacc[c] - (isLast ? t[c] : 0.0);
  pbar[c] += rb / u[c];
  ubar[c] = -rb * r[c] / u[c];
}

__global__ void k_snap_s(const int* done, const double* s3, double* s_out) {
  int i = blockIdx.x * 256 + threadIdx.x;
  if (done[0] == 0) s_out[i] = s3[i];
}

__global__ void k_wstep(const double* __restrict__ r3, const double* __restrict__ t,
                        const double* __restrict__ p, const double* __restrict__ pbar,
                        const double* __restrict__ reg, double* __restrict__ w,
                        double* __restrict__ vmom, double* __restrict__ last_w,
                        int* __restrict__ done, double* __restrict__ r_out,
                        double* __restrict__ loss_out, double* __restrict__ reg_out) {
  int c = threadIdx.x;
  __shared__ double sm[256];
  double loss = -blk_sum(r3[c] * t[c], sm);
  double gamma = 3.0 * loss + 0.05;
  double dot = blk_sum(pbar[c] * p[c], sm);
  double g = p[c] * (pbar[c] - dot) + gamma * (p[c] - 1.0 / (double)CC) / (double)CC;
  double gn = sqrt(blk_sum(g * g, sm));
  g *= fmin(1.0, 1.0 / (gn + 1e-6));
  double dl = p[c] - last_w[c];
  int conv = sqrt(blk_sum(dl * dl, sm)) < 1e-4;
  int wasdone = done[0];
  __syncthreads();
  if (!wasdone) {
    double vm = 0.9 * vmom[c] + g;
    vmom[c] = vm;
    w[c] -= 0.01 * vm;
    last_w[c] = p[c];
    r_out[c] = r3[c];
    if (!c) {
      loss_out[0] = loss;
      reg_out[0] = reg[0];
      if (conv) done[0] = 1;
    }
  }
}

__global__ void k_init(const float* w_in, double* w, double* vmom, double* last_w,
                       int* done, double* s0, double* loss_out, double* reg_out) {
  int i = blockIdx.x * 256 + threadIdx.x;
  if (i < BSZ) s0[i] = 1.0;
  if (i < CC) { w[i] = (double)w_in[i]; vmom[i] = 0.0; last_w[i] = 0.0; }
  if (i == 0) { done[0] = 0; loss_out[0] = 0.0; reg_out[0] = 0.0; }
}

__global__ void k_finalQ(const float* __restrict__ E, const double* __restrict__ r_out,
                         const double* __restrict__ s_out, const double* __restrict__ loss_out,
                         const double* __restrict__ reg_out, float* __restrict__ out) {
  int b = blockIdx.x, c = threadIdx.x;
  out[(size_t)b * CC + c] =
      (float)((double)BSZ * r_out[c] * (double)E[(size_t)b * CC + c] * s_out[b]);
  if (b == 0 && c == 0) {
    out[(size_t)BSZ * CC] = (float)loss_out[0];
    out[(size_t)BSZ * CC + 1] = (float)reg_out[0];
  }
}

// ---------------------------------------------------------------------------
extern "C" void kernel_launch(void* const* d_in, const int* in_sizes, int n_in,
                              void* d_out, int out_size, void* d_ws, size_t ws_size,
                              hipStream_t stream) {
  const float* F = (const float*)d_in[0];   // features 8192x2048
  const float* H = (const float*)d_in[1];   // head     2048x256
  const float* W0 = (const float*)d_in[2];  // w        1x256
  float* out = (float*)d_out;

  // bump allocator over workspace
  char* wsp = (char*)d_ws;
  size_t off = 0;
  auto alloc = [&](size_t bytes) -> void* {
    void* r = wsp + off;
    off += (bytes + 255) & ~(size_t)255;
    return r;
  };
  float* LOGITS = (float*)alloc((size_t)BSZ * CC * 4);
  float* E      = (float*)alloc((size_t)BSZ * CC * 4);
  float* L      = (float*)alloc((size_t)BSZ * CC * 4);
  float* RN     = (float*)alloc(BSZ * 4);
  float* CN     = (float*)alloc(CC * 4);
  double* S[4];
  for (int k = 0; k < 4; k++) S[k] = (double*)alloc(BSZ * 8);
  double* U[3];
  double* R[3];
  for (int k = 0; k < 3; k++) { U[k] = (double*)alloc(CC * 8); R[k] = (double*)alloc(CC * 8); }
  double* P     = (double*)alloc(CC * 8);
  double* PBAR  = (double*)alloc(CC * 8);
  double* T     = (double*)alloc(CC * 8);
  double* CACC  = (double*)alloc(CC * 8);
  double* UBAR  = (double*)alloc(CC * 8);
  double* WST   = (double*)alloc(CC * 8);
  double* VMOM  = (double*)alloc(CC * 8);
  double* LASTW = (double*)alloc(CC * 8);
  double* ROUT  = (double*)alloc(CC * 8);
  double* QV    = (double*)alloc(BSZ * 8);
  double* VBAR  = (double*)alloc(BSZ * 8);
  double* SBAR  = (double*)alloc(BSZ * 8);
  double* SOUT  = (double*)alloc(BSZ * 8);
  double* REG   = (double*)alloc(256);
  double* LOSSO = (double*)alloc(256);
  double* REGO  = (double*)alloc(256);
  int*    DONE  = (int*)alloc(256);
  (void)ws_size; (void)in_sizes; (void)n_in; (void)out_size;

  // setup
  k_init<<<33, 256, 0, stream>>>(W0, WST, VMOM, LASTW, DONE, S[0], LOSSO, REGO);
  k_rownorm<<<BSZ, 256, 0, stream>>>(F, RN);
  k_colnorm<<<CC, 256, 0, stream>>>(H, CN);
  k_gemm<<<dim3(BSZ / 64, 2), 256, 0, stream>>>(F, H, RN, CN, LOGITS);
  k_lse<<<BSZ, 256, 0, stream>>>(LOGITS, E, L);

  for (int it = 0; it < NOUT; it++) {
    // p = softmax(w), reg, zero pbar
    k_pre<<<1, 256, 0, stream>>>(WST, P, PBAR, REG);
    // forward sinkhorn (factored): u=E s; r=p/u; s=1/(B E^T r)
    for (int k = 0; k < 3; k++) {
      k_colmv<<<CC, 256, 0, stream>>>(E, L, 0, S[k], U[k]);
      k_r<<<1, 256, 0, stream>>>(P, U[k], R[k]);
      k_rowmv<<<BSZ / 8, 256, 0, stream>>>(E, L, 0, R[k], S[k + 1], 1);
    }
    // loss contractions: t_c = sum_b E L s3 ; q_b = sum_c E L r3
    k_colmv<<<CC, 256, 0, stream>>>(E, L, 1, S[3], T);
    k_rowmv<<<BSZ / 8, 256, 0, stream>>>(E, L, 1, R[2], QV, 0);
    // backward k=3
    k_vbar<<<BSZ / 256, 256, 0, stream>>>(S[3], QV, +(double)BSZ, VBAR);
    k_colmv<<<CC, 256, 0, stream>>>(E, L, 0, VBAR, CACC);
    k_bkc<<<1, 256, 0, stream>>>(CACC, T, 1, U[2], R[2], PBAR, UBAR);
    k_rowmv<<<BSZ / 8, 256, 0, stream>>>(E, L, 0, UBAR, SBAR, 0);
    // backward k=2
    k_vbar<<<BSZ / 256, 256, 0, stream>>>(S[2], SBAR, -(double)BSZ, VBAR);
    k_colmv<<<CC, 256, 0, stream>>>(E, L, 0, VBAR, CACC);
    k_bkc<<<1, 256, 0, stream>>>(CACC, T, 0, U[1], R[1], PBAR, UBAR);
    k_rowmv<<<BSZ / 8, 256, 0, stream>>>(E, L, 0, UBAR, SBAR, 0);
    // backward k=1 (s0 is constant; no further propagation)
    k_vbar<<<BSZ / 256, 256, 0, stream>>>(S[1], SBAR, -(double)BSZ, VBAR);
    k_colmv<<<CC, 256, 0, stream>>>(E, L, 0, VBAR, CACC);
    k_bkc<<<1, 256, 0, stream>>>(CACC, T, 0, U[0], R[0], PBAR, UBAR);
    // snapshots (guarded on pre-update done) + w update
    k_snap_s<<<BSZ / 256, 256, 0, stream>>>(DONE, S[3], SOUT);
    k_wstep<<<1, 256, 0, stream>>>(R[2], T, P, PBAR, REG, WST, VMOM, LASTW, DONE,
                                   ROUT, LOSSO, REGO);
  }

  // Q[b,c] = B * r_c * E[b,c] * s_b ; then loss, reg
  k_finalQ<<<BSZ, 256, 0, stream>>>(E, ROUT, SOUT, LOSSO, REGO, out);
}